// FRAP3_47047071761013
// MI455X (gfx1250) — compile-verified
//
#include <hip/hip_runtime.h>

// MI455X / gfx1250, wave32. Single-wave latency-optimized FRAP3 kernel.
// All MLP matmuls run on V_WMMA_F32_16X16X4_F32 (full fp32 reference precision).
//
// Strategy: one workgroup = one wave (32 threads). Zero the LDS arena, stage all
// weights TRANSPOSED + ZERO-PADDED into LDS, stage biases zero-padded. Then every
// WMMA operand is a single unguarded ds_load_b64 and every tile store is 8
// unconditional ds_store_b32 (padded columns self-compute to 0), eliminating the
// exec-mask branch/wait overhead that dominated the previous version.
//
// d_in flattening (setup_inputs dict order):
//  [0] waiting(8) [1] phase(8)
//  cars:   [2] W1(1x12)  [3] b1(12) [4] W2(12x4)  [5] b2(4)
//  phase:  [6] W1(8x64)  [7] b1(64) [8] W2(64x16) [9] b2(16) [10] W3(16x4) [11] b3(4)
//  demand: [12] W1(8x32) [13] b1(32)[14] W2(32x16)[15] b2(16)[16] W3(16x4) [17] b3(4)
//  convD:  [18] W1(8x4)  [19] b1(4) [20] W2(4x2)  [21] b2(2)
//  convE:  [22] W1(1x8)  [23] b1(8) [24] W2(8x2)  [25] b2(2)
//  convC:  [26] W1(2x4)  [27] b1(4) [28] W2(4x1)  [29] b2(1)

typedef float v2f __attribute__((ext_vector_type(2)));
typedef float v4f __attribute__((ext_vector_type(4)));
typedef float v8f __attribute__((ext_vector_type(8)));

struct FrapParams {
  const float *waiting, *phase;
  const float *cW1, *cB1, *cW2, *cB2;
  const float *pW1, *pB1, *pW2, *pB2, *pW3, *pB3;
  const float *dW1, *dB1, *dW2, *dB2, *dW3, *dB3;
  const float *DW1, *DB1, *DW2, *DB2;
  const float *EW1, *EB1, *EW2, *EB2;
  const float *CW1, *CB1, *CW2, *CB2;
  float *out;
};

// ---- LDS arena layout (float offsets; all multiples of 4 -> 16B aligned) ----
// Transposed weights Bt[n][k] (N padded to 16, K padded to mult of 4):
#define OFF_PW1T 0      // [64][8]
#define OFF_PW2T 512    // [16][64]
#define OFF_PW3T 1536   // [16][16]
#define OFF_CW1T 1792   // [16][4]
#define OFF_CW2T 1856   // [16][16]
#define OFF_DW1T 2112   // [32][8]
#define OFF_DW2T 2368   // [16][32]
#define OFF_DW3T 2880   // [16][16]
#define OFF_BIAS 3136   // 192 floats of zero-padded biases
#define OFF_AH1  3328   // [16][64] phase hidden 1
#define OFF_AH2  4352   // [16][16] phase hidden 2
#define OFF_ANPH 4608   // [16][16] n_phase (row0 cols0..3 valid)
#define OFF_AM1  4864   // [16][16] cars hidden
#define OFF_ACARS 5120  // [16][16] cars out
#define OFF_AX   5376   // [16][8]  demand input
#define OFF_AHD1 5504   // [16][32] demand hidden 1
#define OFF_AHD2 6016   // [16][16] demand hidden 2
#define OFF_ADEM 6272   // [16][16] demand out
#define OFF_XPH  6528   // [16] phase input row (A lda=8, over-read stays in arena)
#define OFF_WBUF 6544   // [8][4] cars input (K padded 1->4, zeros from arena clear)
#define OFF_DPV  6576   // [8][4] demand_phase
#define OFF_CPOS 6608   // [64] conv output per (i,j)
#define SMEM_FLOATS 6672

// bias sub-offsets within OFF_BIAS
#define B_CB1 0
#define B_CB2 16
#define B_PB1 32
#define B_PB2 96
#define B_PB3 112
#define B_DB1 128
#define B_DB2 160
#define B_DB3 176

__device__ inline v8f wmma_acc(v2f a, v2f b, v8f c) {
  // 8 args: (neg_a, A, neg_b, B, c_mod, C, reuse_a, reuse_b)
  return __builtin_amdgcn_wmma_f32_16x16x4_f32(false, a, false, b, (short)0, c,
                                               false, false);
}

// C(16x16) = A(16xK) * Bt(16-col tile at ncol0), all operands in LDS,
// K multiple of 4. Fragment layouts per CDNA5 ISA 7.12.2:
//   A: lane l holds row l, K offsets {0,1} (lanes 0-15) / {2,3} (lanes 16-31)
//   B: lane l holds col ncol0+l, same K split. Both fetches are ds_load_b64.
__device__ inline v8f gemm_lds(const float* A, int lda, const float* Bt, int ldb,
                               int K, int ncol0, int lane) {
  const int half = lane >> 4;
  const int l = lane & 15;
  const float* ap = A + l * lda + 2 * half;
  const float* bp = Bt + (ncol0 + l) * ldb + 2 * half;
  v8f c = {0.f, 0.f, 0.f, 0.f, 0.f, 0.f, 0.f, 0.f};
  for (int kk = 0; kk < K; kk += 4) {
    v2f a = *(const v2f*)(ap + kk);   // 8B-aligned (lda even, offsets even)
    v2f b = *(const v2f*)(bp + kk);
    c = wmma_acc(a, b, c);
  }
  return c;
}

// Unconditional full 16x16 tile store with bias + optional relu. Padded columns
// hold 0 (B and bias zero-padded), rows >= real M hold garbage that is never
// consumed (A row m only affects C row m).
__device__ inline void store_tile16(float* dst, int ldd, v8f c, int lane,
                                    const float* blds, int bcol0, bool relu) {
  const int half = lane >> 4;
  const int l = lane & 15;
  const float bv = blds[bcol0 + l];
#pragma unroll
  for (int r = 0; r < 8; ++r) {
    float v = c[r] + bv;
    if (relu) v = fmaxf(v, 0.f);
    dst[(r + 8 * half) * ldd + l] = v;
  }
}

// Fill transposed weight cache: bt[n*Kpad + k] = W[k*N + n] (pad stays zero).
__device__ inline void stage_w(float* bt, const float* W, int K, int N, int Kpad,
                               int tid) {
  for (int i = tid; i < K * N; i += 32) bt[(i % N) * Kpad + (i / N)] = W[i];
}

__device__ inline void stage_c(float* dst, const float* src, int n, int tid) {
  for (int i = tid; i < n; i += 32) dst[i] = src[i];
}

__global__ __launch_bounds__(32) void frap3_kernel(FrapParams p) {
  __shared__ __align__(16) float smem[SMEM_FLOATS];
  const int tid = threadIdx.x;

  // ---- clear entire arena (vectorized), then stage weights/biases/inputs ----
  {
    v4f z = {0.f, 0.f, 0.f, 0.f};
    v4f* s4 = (v4f*)smem;
    for (int i = tid; i < SMEM_FLOATS / 4; i += 32) s4[i] = z;
  }
  stage_w(smem + OFF_PW1T, p.pW1, 8, 64, 8, tid);
  stage_w(smem + OFF_PW2T, p.pW2, 64, 16, 64, tid);
  stage_w(smem + OFF_PW3T, p.pW3, 16, 4, 16, tid);
  stage_w(smem + OFF_CW1T, p.cW1, 1, 12, 4, tid);
  stage_w(smem + OFF_CW2T, p.cW2, 12, 4, 16, tid);
  stage_w(smem + OFF_DW1T, p.dW1, 8, 32, 8, tid);
  stage_w(smem + OFF_DW2T, p.dW2, 32, 16, 32, tid);
  stage_w(smem + OFF_DW3T, p.dW3, 16, 4, 16, tid);
  float* bias = smem + OFF_BIAS;
  stage_c(bias + B_CB1, p.cB1, 12, tid);
  stage_c(bias + B_CB2, p.cB2, 4, tid);
  stage_c(bias + B_PB1, p.pB1, 64, tid);
  stage_c(bias + B_PB2, p.pB2, 16, tid);
  stage_c(bias + B_PB3, p.pB3, 4, tid);
  stage_c(bias + B_DB1, p.dB1, 32, tid);
  stage_c(bias + B_DB2, p.dB2, 16, tid);
  stage_c(bias + B_DB3, p.dB3, 4, tid);
  if (tid < 8) smem[OFF_XPH + tid] = p.phase[tid];
  if (tid < 8) smem[OFF_WBUF + tid * 4] = p.waiting[tid];  // col0; cols1..3 zero
  __syncthreads();

  // ---- phase MLP: [1,8] -> 64 -> 16 -> 4 (only row 0 meaningful) ----
  for (int t = 0; t < 4; ++t) {
    v8f c = gemm_lds(smem + OFF_XPH, 8, smem + OFF_PW1T, 8, 8, 16 * t, tid);
    store_tile16(smem + OFF_AH1 + 16 * t, 64, c, tid, bias + B_PB1, 16 * t, true);
  }
  __syncthreads();
  {
    v8f c = gemm_lds(smem + OFF_AH1, 64, smem + OFF_PW2T, 64, 64, 0, tid);
    store_tile16(smem + OFF_AH2, 16, c, tid, bias + B_PB2, 0, true);
  }
  __syncthreads();
  {
    v8f c = gemm_lds(smem + OFF_AH2, 16, smem + OFF_PW3T, 16, 16, 0, tid);
    store_tile16(smem + OFF_ANPH, 16, c, tid, bias + B_PB3, 0, false);
  }

  // ---- cars MLP: [8,1] -> 12 -> 4 (rows 0..7 meaningful) ----
  {
    v8f c = gemm_lds(smem + OFF_WBUF, 4, smem + OFF_CW1T, 4, 4, 0, tid);
    store_tile16(smem + OFF_AM1, 16, c, tid, bias + B_CB1, 0, true);
  }
  __syncthreads();
  {
    v8f c = gemm_lds(smem + OFF_AM1, 16, smem + OFF_CW2T, 16, 16, 0, tid);
    store_tile16(smem + OFF_ACARS, 16, c, tid, bias + B_CB2, 0, false);
  }
  __syncthreads();
  {  // X = [cars | n_phase] rows 0..7 (32 lanes = 8 rows x 4 cols)
    const int r = tid >> 2, cc = tid & 3;
    smem[OFF_AX + r * 8 + cc] = smem[OFF_ACARS + r * 16 + cc];
    smem[OFF_AX + r * 8 + 4 + cc] = smem[OFF_ANPH + cc];
  }
  __syncthreads();

  // ---- demand MLP: [8,8] -> 32 -> 16 -> 4 ----
  for (int t = 0; t < 2; ++t) {
    v8f c = gemm_lds(smem + OFF_AX, 8, smem + OFF_DW1T, 8, 8, 16 * t, tid);
    store_tile16(smem + OFF_AHD1 + 16 * t, 32, c, tid, bias + B_DB1, 16 * t, true);
  }
  __syncthreads();
  {
    v8f c = gemm_lds(smem + OFF_AHD1, 32, smem + OFF_DW2T, 32, 32, 0, tid);
    store_tile16(smem + OFF_AHD2, 16, c, tid, bias + B_DB2, 0, true);
  }
  __syncthreads();
  {
    v8f c = gemm_lds(smem + OFF_AHD2, 16, smem + OFF_DW3T, 16, 16, 0, tid);
    store_tile16(smem + OFF_ADEM, 16, c, tid, bias + B_DB3, 0, false);
  }
  __syncthreads();

  // ---- demand_phase: dp[p] = demand[P0[p]] + demand[P1[p]] ----
  {
    const int pp = tid >> 2, cc = tid & 3;
    const int i0 = pp & 3;
    const int i1 = (pp < 4) ? (pp + 4) : (pp ^ 1);  // POSSIBLE second lane
    smem[OFF_DPV + tid] =
        smem[OFF_ADEM + i0 * 16 + cc] + smem[OFF_ADEM + i1 * 16 + cc];
  }
  __syncthreads();

  // ---- conv blocks: one (i,j) position per lane, 56 positions in 2 passes ----
  for (int pass = 0; pass < 2; ++pass) {
    const int pos = pass * 32 + tid;
    if (pos < 56) {
      const int i = pos / 7, j = pos % 7;
      const float m = (i != j) ? 1.f : 0.f;

      float dch[8];
#pragma unroll
      for (int c = 0; c < 4; ++c) {
        dch[c]     = smem[OFF_DPV + i * 4 + c] * m;  // D_top channels
        dch[4 + c] = smem[OFF_DPV + j * 4 + c] * m;  // D_bot channels
      }

      // convD: 8 -> 4 (relu) -> 2
      float hdA = p.DB2[0], hdB = p.DB2[1];
#pragma unroll
      for (int o = 0; o < 4; ++o) {
        float s = p.DB1[o];
#pragma unroll
        for (int c = 0; c < 8; ++c) s += dch[c] * p.DW1[c * 4 + o];
        s = fmaxf(s, 0.f);
        hdA += s * p.DW2[o * 2 + 0];
        hdB += s * p.DW2[o * 2 + 1];
      }

      // convE on COMP constant: comp(i,j) via bit trick on competition pairs
      const int jj = (j < i) ? j : (j + 1);
      const int mn = (i < jj) ? i : jj;
      const int mx = (i < jj) ? jj : i;
      const float e = (((mx >> 1) - (mn >> 1)) == 2) ? 1.f : 0.f;
      float hrA = p.EB2[0], hrB = p.EB2[1];
#pragma unroll
      for (int o = 0; o < 8; ++o) {
        const float s = fmaxf(e * p.EW1[o] + p.EB1[o], 0.f);
        hrA += s * p.EW2[o * 2 + 0];
        hrB += s * p.EW2[o * 2 + 1];
      }

      // H_c = H_d * H_r ; convC: 2 -> 4 (relu) -> 1
      const float hc0 = hdA * hrA, hc1 = hdB * hrB;
      float acc = p.CB2[0];
#pragma unroll
      for (int o = 0; o < 4; ++o) {
        const float s =
            fmaxf(hc0 * p.CW1[o] + hc1 * p.CW1[4 + o] + p.CB1[o], 0.f);
        acc += s * p.CW2[o];
      }
      smem[OFF_CPOS + pos] = acc;
    }
  }
  __syncthreads();

  // ---- reduce over j: out[i] = sum_j C[0,i,j] ----
  if (tid < 8) {
    float s = 0.f;
#pragma unroll
    for (int j = 0; j < 7; ++j) s += smem[OFF_CPOS + tid * 7 + j];
    p.out[tid] = s;
  }
}

extern "C" void kernel_launch(void* const* d_in, const int* in_sizes, int n_in,
                              void* d_out, int out_size, void* d_ws,
                              size_t ws_size, hipStream_t stream) {
  (void)in_sizes; (void)n_in; (void)d_ws; (void)ws_size; (void)out_size;
  FrapParams p;
  p.waiting = (const float*)d_in[0];
  p.phase   = (const float*)d_in[1];
  p.cW1 = (const float*)d_in[2];  p.cB1 = (const float*)d_in[3];
  p.cW2 = (const float*)d_in[4];  p.cB2 = (const float*)d_in[5];
  p.pW1 = (const float*)d_in[6];  p.pB1 = (const float*)d_in[7];
  p.pW2 = (const float*)d_in[8];  p.pB2 = (const float*)d_in[9];
  p.pW3 = (const float*)d_in[10]; p.pB3 = (const float*)d_in[11];
  p.dW1 = (const float*)d_in[12]; p.dB1 = (const float*)d_in[13];
  p.dW2 = (const float*)d_in[14]; p.dB2 = (const float*)d_in[15];
  p.dW3 = (const float*)d_in[16]; p.dB3 = (const float*)d_in[17];
  p.DW1 = (const float*)d_in[18]; p.DB1 = (const float*)d_in[19];
  p.DW2 = (const float*)d_in[20]; p.DB2 = (const float*)d_in[21];
  p.EW1 = (const float*)d_in[22]; p.EB1 = (const float*)d_in[23];
  p.EW2 = (const float*)d_in[24]; p.EB2 = (const float*)d_in[25];
  p.CW1 = (const float*)d_in[26]; p.CB1 = (const float*)d_in[27];
  p.CW2 = (const float*)d_in[28]; p.CB2 = (const float*)d_in[29];
  p.out = (float*)d_out;
  frap3_kernel<<<1, 32, 0, stream>>>(p);
}